// NEQUIPLayer_35098472743024
// MI455X (gfx1250) — compile-verified
//
#include <hip/hip_runtime.h>
#include <math.h>

// ---------------------------------------------------------------------------
// NequIP layer for MI455X (gfx1250), wave32, f32 WMMA 16x16x4 + TDM staging.
// ---------------------------------------------------------------------------

typedef __attribute__((ext_vector_type(2))) float v2f;
typedef __attribute__((ext_vector_type(8))) float v8f;
typedef __attribute__((ext_vector_type(4))) unsigned int tdm_g0_t;
typedef __attribute__((ext_vector_type(8))) int tdm_g1_t;
typedef __attribute__((ext_vector_type(4))) int tdm_g2_t;
typedef __attribute__((ext_vector_type(8))) int tdm_g4_t;

__device__ inline v8f zero8() {
  v8f z = {0.f, 0.f, 0.f, 0.f, 0.f, 0.f, 0.f, 0.f};
  return z;
}

__device__ inline v8f wmma4(v2f a, v2f b, v8f c) {
  // D = A(16x4) x B(4x16) + C(16x16), fp32
  return __builtin_amdgcn_wmma_f32_16x16x4_f32(
      /*neg_a=*/false, a, /*neg_b=*/false, b,
      /*c_mod=*/(short)0, c, /*reuse_a=*/false, /*reuse_b=*/false);
}

__device__ inline float swish_f(float x) { return x / (1.f + __expf(-x)); }

// A fragment, strided k: lane holds row m, k = k0 + (lane>>4)*2 + {0,1}.
__device__ inline v2f afrag_row(const float* rowptr, int k0, int estr,
                                int lane) {
  int kb = k0 + ((lane >> 4) << 1);
  v2f a;
  a.x = rowptr[(size_t)kb * estr];
  a.y = rowptr[(size_t)(kb + 1) * estr];
  return a;
}

// A fragment, unit-stride k: single 64-bit load (rowptr 8B-aligned, kb even).
__device__ inline v2f afrag_row2(const float* rowptr, int k0, int lane) {
  int kb = k0 + ((lane >> 4) << 1);
  return *(const v2f*)(rowptr + kb);
}

// B fragment from row-major W[K][ld]: lane holds col n = n0 + (lane&15),
// k = k0 + (lane>>4)*2 + {0,1}.
__device__ inline v2f bfrag(const float* W, int ld, int k0, int n0, int lane) {
  int n = n0 + (lane & 15);
  int kb = k0 + ((lane >> 4) << 1);
  v2f b;
  b.x = W[(size_t)kb * ld + n];
  b.y = W[(size_t)(kb + 1) * ld + n];
  return b;
}

__device__ inline void wait_tensor0() {
#if __has_builtin(__builtin_amdgcn_s_wait_tensorcnt)
  __builtin_amdgcn_s_wait_tensorcnt(0);
#else
  asm volatile("s_wait_tensorcnt 0x0" ::: "memory");
#endif
}

// ---------------------------------------------------------------------------
// Kernel 0: zero the aggregation buffers.
// ---------------------------------------------------------------------------
__global__ void zero_kernel(float* __restrict__ p, size_t n) {
  size_t i = (size_t)blockIdx.x * blockDim.x + threadIdx.x;
  size_t stride = (size_t)gridDim.x * blockDim.x;
  for (; i < n; i += stride) p[i] = 0.f;
}

// ---------------------------------------------------------------------------
// Kernel 1: node up-projection + species-indexed skip connection (all WMMA).
//   h_s [N,64], h_v [N,64,3], sc_s [N,128], sc_v [N,64,3]
// One 16-node tile per wave, 4 waves (64 nodes) per block.
// ---------------------------------------------------------------------------
__global__ __launch_bounds__(128) void node_up_kernel(
    const float* __restrict__ node_s, const float* __restrict__ node_v,
    const int* __restrict__ specie, const float* __restrict__ W_up_s,
    const float* __restrict__ W_up_v, const float* __restrict__ W_skip_s,
    const float* __restrict__ W_skip_v, float* __restrict__ h_s,
    float* __restrict__ h_v, float* __restrict__ sc_s,
    float* __restrict__ sc_v, int N) {
  const int lane = threadIdx.x & 31;
  const int wave = threadIdx.x >> 5;
  const int m0 = blockIdx.x * 64 + wave * 16;
  if (m0 >= N) return;

  const float S64 = 0.125f;  // 1/sqrt(64)
  const int nloc = lane & 15;
  const int mb8 = (lane >> 4) << 3;
  const bool full = (m0 + 16 <= N);
  const int mrow = m0 + (lane & 15);
  const int mrowc = (mrow < N) ? mrow : (N - 1);
  const int sp = specie[mrowc];
  const float* As = node_s + (size_t)mrowc * 64;

  // ---- h_s = node_s @ W_up_s * S64  (K=64, 4 N-tiles)
  for (int nt = 0; nt < 4; ++nt) {
    v8f acc = zero8();
    for (int ks = 0; ks < 16; ++ks) {
      v2f a = afrag_row2(As, ks * 4, lane);
      v2f b = bfrag(W_up_s, 64, ks * 4, nt * 16, lane);
      acc = wmma4(a, b, acc);
    }
    if (full) {
      for (int v = 0; v < 8; ++v)
        h_s[(size_t)(m0 + mb8 + v) * 64 + nt * 16 + nloc] = acc[v] * S64;
    } else {
      for (int v = 0; v < 8; ++v) {
        int m = m0 + mb8 + v;
        if (m < N) h_s[(size_t)m * 64 + nt * 16 + nloc] = acc[v] * S64;
      }
    }
  }

  // ---- h_v (per component)
  for (int c = 0; c < 3; ++c) {
    const float* Av = node_v + (size_t)mrowc * 192 + c;
    for (int nt = 0; nt < 4; ++nt) {
      v8f acc = zero8();
      for (int ks = 0; ks < 16; ++ks) {
        v2f a = afrag_row(Av, ks * 4, 3, lane);
        v2f b = bfrag(W_up_v, 64, ks * 4, nt * 16, lane);
        acc = wmma4(a, b, acc);
      }
      if (full) {
        for (int v = 0; v < 8; ++v)
          h_v[(size_t)(m0 + mb8 + v) * 192 + (nt * 16 + nloc) * 3 + c] =
              acc[v] * S64;
      } else {
        for (int v = 0; v < 8; ++v) {
          int m = m0 + mb8 + v;
          if (m < N)
            h_v[(size_t)m * 192 + (nt * 16 + nloc) * 3 + c] = acc[v] * S64;
        }
      }
    }
  }

  // ---- sc_s: species-masked GEMM, accumulate over 5 species (N=128)
  for (int nt = 0; nt < 8; ++nt) {
    v8f acc = zero8();
    for (int s = 0; s < 5; ++s) {
      float msk = (sp == s) ? 1.f : 0.f;
      const float* Ws = W_skip_s + (size_t)s * 64 * 128;
      for (int ks = 0; ks < 16; ++ks) {
        v2f a = afrag_row2(As, ks * 4, lane);
        a.x *= msk;
        a.y *= msk;
        v2f b = bfrag(Ws, 128, ks * 4, nt * 16, lane);
        acc = wmma4(a, b, acc);
      }
    }
    if (full) {
      for (int v = 0; v < 8; ++v)
        sc_s[(size_t)(m0 + mb8 + v) * 128 + nt * 16 + nloc] = acc[v] * S64;
    } else {
      for (int v = 0; v < 8; ++v) {
        int m = m0 + mb8 + v;
        if (m < N) sc_s[(size_t)m * 128 + nt * 16 + nloc] = acc[v] * S64;
      }
    }
  }

  // ---- sc_v: species-masked GEMM per component (N=64)
  for (int c = 0; c < 3; ++c) {
    const float* Av = node_v + (size_t)mrowc * 192 + c;
    for (int nt = 0; nt < 4; ++nt) {
      v8f acc = zero8();
      for (int s = 0; s < 5; ++s) {
        float msk = (sp == s) ? 1.f : 0.f;
        const float* Wv = W_skip_v + (size_t)s * 64 * 64;
        for (int ks = 0; ks < 16; ++ks) {
          v2f a = afrag_row(Av, ks * 4, 3, lane);
          a.x *= msk;
          a.y *= msk;
          v2f b = bfrag(Wv, 64, ks * 4, nt * 16, lane);
          acc = wmma4(a, b, acc);
        }
      }
      if (full) {
        for (int v = 0; v < 8; ++v)
          sc_v[(size_t)(m0 + mb8 + v) * 192 + (nt * 16 + nloc) * 3 + c] =
              acc[v] * S64;
      } else {
        for (int v = 0; v < 8; ++v) {
          int m = m0 + mb8 + v;
          if (m < N)
            sc_v[(size_t)m * 192 + (nt * 16 + nloc) * 3 + c] = acc[v] * S64;
        }
      }
    }
  }
}

// ---------------------------------------------------------------------------
// Kernel 2: fused edge kernel. 64 threads / 32 edges per block.
//   basis -> WMMA MLP (rb->h1->h2->mix in LDS) -> gather/TP/scale/scatter.
// W2 (16 KB) is DMA'd once per block into the first 16 KB of s_mix (dead
// until layer 3) by the Tensor Data Mover; W3 stays in global on purpose:
// staging it would cost 64 KB LDS and drop occupancy from ~12 to ~4 waves
// per WGP, which the atomic-scatter phase cannot afford.
// ---------------------------------------------------------------------------
__global__ __launch_bounds__(64) void edge_kernel(
    const float* __restrict__ vectors, const int* __restrict__ senders,
    const int* __restrict__ receivers, const float* __restrict__ W1,
    const float* __restrict__ W2, const float* __restrict__ W3,
    const float* __restrict__ h_s, const float* __restrict__ h_v,
    float* __restrict__ agg_s, float* __restrict__ agg_v, int E) {
  __shared__ float s_rb[32 * 8];     //  1 KB
  __shared__ float s_h1[32 * 64];    //  8 KB
  __shared__ float s_h2[32 * 64];    //  8 KB
  __shared__ float s_mix[32 * 256];  // 32 KB (first 16 KB aliased as W2)
  __shared__ float s_y[32 * 3];      // unit vectors

  const int tid = threadIdx.x;
  const int lane = tid & 31;
  const int wave = tid >> 5;
  const long long e0 = (long long)blockIdx.x * 32;
  float* s_w2 = s_mix;  // W2 staged here until layer 3 overwrites it

  // --- phase 1: geometry + radial basis (one edge per thread 0..31)
  if (tid < 32) {
    long long e = e0 + tid;
    float vx = 0.f, vy = 0.f, vz = 0.f;
    if (e < (long long)E) {
      vx = vectors[e * 3 + 0];
      vy = vectors[e * 3 + 1];
      vz = vectors[e * 3 + 2];
    }
    float len = sqrtf(vx * vx + vy * vy + vz * vz);
    float sl = (len == 0.f) ? 1.f : len;
    float inv = 1.f / sl;
    s_y[tid * 3 + 0] = vx * inv;
    s_y[tid * 3 + 1] = vy * inv;
    s_y[tid * 3 + 2] = vz * inv;
    // poly_envelope(5,2): 1 - 28 x^6 + 48 x^7 - 21 x^8, zero past 1
    float x2 = len * len, x3 = x2 * len, x6 = x3 * x3, x7 = x6 * len,
          x8 = x7 * len;
    float env = (len < 1.f) ? (1.f - 28.f * x6 + 48.f * x7 - 21.f * x8) : 0.f;
    float pref = ((len == 0.f) || (e >= (long long)E))
                     ? 0.f
                     : 1.41421356237f * inv * env;  // sqrt(2)/len * env
    const float PI = 3.14159265358979f;
    for (int k = 0; k < 8; ++k)
      s_rb[tid * 8 + k] = pref * __sinf((float)(k + 1) * PI * sl);
  }

  // --- stage W2 [64x64] f32 into LDS via the Tensor Data Mover.
#if __has_builtin(__builtin_amdgcn_tensor_load_to_lds)
  if (wave == 0) {
    // D# per CDNA5 ISA: 1 row x 4096 elems, 4B each.
    unsigned lds_addr = (unsigned)(size_t)(void*)s_w2;
    unsigned long long ga = (unsigned long long)(size_t)(const void*)W2;
    tdm_g0_t g0;
    g0.x = 1u;                   // count=1 (valid user descriptor)
    g0.y = lds_addr;             // lds_addr[31:0]
    g0.z = (unsigned)ga;         // global_addr[31:0]
    g0.w = (unsigned)((ga >> 32) & 0x01FFFFFFu) | (2u << 30);  // [56:32]|type=2
    tdm_g1_t g1;
    g1[0] = 2 << 16;             // data_size = 4 bytes
    g1[1] = (int)(4096u << 16);  // tensor_dim0 = 4096 (low16 @ bits[63:48])
    g1[2] = 1 << 16;             // tensor_dim1 = 1    (low16 @ bits[95:80])
    g1[3] = (int)(4096u << 16);  // tile_dim0 = 4096   (bits[127:112])
    g1[4] = 1;                   // tile_dim1 = 1, tile_dim2 = 0
    g1[5] = 4096;                // tensor_dim0_stride = 4096
    g1[6] = 0;
    g1[7] = 0;
    tdm_g2_t gz;
    gz[0] = 0; gz[1] = 0; gz[2] = 0; gz[3] = 0;
    tdm_g4_t gz8;
    gz8[0] = 0; gz8[1] = 0; gz8[2] = 0; gz8[3] = 0;
    gz8[4] = 0; gz8[5] = 0; gz8[6] = 0; gz8[7] = 0;
    // 6-arg form (clang-23 / therock-10.0 headers)
    __builtin_amdgcn_tensor_load_to_lds(g0, g1, gz, gz, gz8, 0);
    wait_tensor0();
  }
#else
  for (int i = tid; i < 64 * 64; i += 64) s_w2[i] = W2[i];
#endif
  __syncthreads();

  // --- phase 2: MLP via WMMA, one 16-edge tile per wave
  const int m0 = wave * 16;
  const int nloc = lane & 15;
  const int mb8 = (lane >> 4) << 3;
  const float INV8 = 0.353553390593f;  // 1/sqrt(8)
  const float INV64 = 0.125f;

  const float* Arb = &s_rb[(m0 + (lane & 15)) * 8];
  for (int nt = 0; nt < 4; ++nt) {  // layer 1: [16x8]@[8x64]
    v8f acc = zero8();
    for (int ks = 0; ks < 2; ++ks) {
      v2f a = afrag_row2(Arb, ks * 4, lane);
      v2f b = bfrag(W1, 64, ks * 4, nt * 16, lane);
      acc = wmma4(a, b, acc);
    }
    for (int v = 0; v < 8; ++v)
      s_h1[(m0 + mb8 + v) * 64 + nt * 16 + nloc] = swish_f(acc[v] * INV8);
  }
  const float* A1 = &s_h1[(m0 + (lane & 15)) * 64];
  for (int nt = 0; nt < 4; ++nt) {  // layer 2: [16x64]@[64x64], W2 from LDS
    v8f acc = zero8();
    for (int ks = 0; ks < 16; ++ks) {
      v2f a = afrag_row2(A1, ks * 4, lane);
      v2f b = bfrag(s_w2, 64, ks * 4, nt * 16, lane);
      acc = wmma4(a, b, acc);
    }
    for (int v = 0; v < 8; ++v)
      s_h2[(m0 + mb8 + v) * 64 + nt * 16 + nloc] = swish_f(acc[v] * INV64);
  }
  __syncthreads();  // all waves done reading W2 before s_mix is overwritten
  const float* A2 = &s_h2[(m0 + (lane & 15)) * 64];
  for (int nt = 0; nt < 16; ++nt) {  // layer 3: [16x64]@[64x256]
    v8f acc = zero8();
    for (int ks = 0; ks < 16; ++ks) {
      v2f a = afrag_row2(A2, ks * 4, lane);
      v2f b = bfrag(W3, 256, ks * 4, nt * 16, lane);
      acc = wmma4(a, b, acc);
    }
    for (int v = 0; v < 8; ++v)
      s_mix[(m0 + mb8 + v) * 256 + nt * 16 + nloc] = acc[v] * INV64;
  }
  __syncthreads();

  // --- phase 3: gather sender features, CG tensor product, scale by mix,
  //     scatter-add to receivers. 512 output atoms per edge:
  //     r<64: m_s ; 64<=r<128: dot(m_v, vhat) ; r>=128: vector part.
  const float SQRT3 = 1.73205080757f;
  const int total = 32 * 512;
  for (int idx = tid; idx < total; idx += 64) {
    int el = idx >> 9;
    int r = idx & 511;
    long long e = e0 + el;
    if (e >= (long long)E) break;  // el is nondecreasing in idx
    int snd = senders[e];
    int rcv = receivers[e];
    const float* mixrow = &s_mix[el * 256];
    if (r < 64) {
      float val = h_s[(size_t)snd * 64 + r] * mixrow[r];
      atomicAdd(&agg_s[(size_t)rcv * 128 + r], val);
    } else if (r < 128) {
      int f = r - 64;
      const float* hv = h_v + (size_t)snd * 192 + f * 3;
      float d = hv[0] * s_y[el * 3 + 0] + hv[1] * s_y[el * 3 + 1] +
                hv[2] * s_y[el * 3 + 2];
      atomicAdd(&agg_s[(size_t)rcv * 128 + r], d * mixrow[r]);
    } else {
      int q = r - 128;    // 0..383
      int f = q / 3;      // vector feature 0..127
      int c = q - f * 3;  // component
      float g = mixrow[128 + f];
      float m;
      if (f < 64)
        m = h_v[(size_t)snd * 192 + q];  // m_v part
      else
        m = h_s[(size_t)snd * 64 + (f - 64)] * (SQRT3 * s_y[el * 3 + c]);
      atomicAdd(&agg_v[(size_t)rcv * 384 + q], m * g);
    }
  }
}

// ---------------------------------------------------------------------------
// Kernel 3: down-projection + skip + gate (WMMA). Gate tile for d_v tile nt
// is d_s tile nt+4 -> identical (lane, vgpr) mapping, pure register gating.
// ---------------------------------------------------------------------------
__global__ __launch_bounds__(128) void node_down_kernel(
    const float* __restrict__ agg_s, const float* __restrict__ agg_v,
    const float* __restrict__ sc_s, const float* __restrict__ sc_v,
    const float* __restrict__ W_down_s, const float* __restrict__ W_down_v,
    float* __restrict__ out, int N) {
  const int lane = threadIdx.x & 31;
  const int wave = threadIdx.x >> 5;
  const int m0 = blockIdx.x * 64 + wave * 16;
  if (m0 >= N) return;

  // 1/sqrt(128) * 1/sqrt(avg_neighbors=12)
  const float CS = 0.0883883476483184f * 0.2886751345948129f;
  const int nloc = lane & 15;
  const int mb8 = (lane >> 4) << 3;
  const bool full = (m0 + 16 <= N);
  const int mrow = m0 + (lane & 15);
  const int mrowc = (mrow < N) ? mrow : (N - 1);
  const float* As = agg_s + (size_t)mrowc * 128;

  v8f gate[4];

  // ---- d_s (K=128, 8 N-tiles). Tiles 0..3 -> scalar outputs; 4..7 -> gates.
  for (int nt = 0; nt < 8; ++nt) {
    v8f acc = zero8();
    for (int ks = 0; ks < 32; ++ks) {
      v2f a = afrag_row2(As, ks * 4, lane);
      v2f b = bfrag(W_down_s, 128, ks * 4, nt * 16, lane);
      acc = wmma4(a, b, acc);
    }
    if (nt < 4) {
      if (full) {
        for (int v = 0; v < 8; ++v) {
          int m = m0 + mb8 + v;
          float ds = acc[v] * CS + sc_s[(size_t)m * 128 + nt * 16 + nloc];
          out[(size_t)m * 256 + nt * 16 + nloc] = swish_f(ds);
        }
      } else {
        for (int v = 0; v < 8; ++v) {
          int m = m0 + mb8 + v;
          if (m < N) {
            float ds = acc[v] * CS + sc_s[(size_t)m * 128 + nt * 16 + nloc];
            out[(size_t)m * 256 + nt * 16 + nloc] = swish_f(ds);
          }
        }
      }
    } else {
      for (int v = 0; v < 8; ++v) {
        int m = m0 + mb8 + v;
        int mc = (m < N) ? m : (N - 1);
        float ds = acc[v] * CS + sc_s[(size_t)mc * 128 + nt * 16 + nloc];
        gate[nt - 4][v] = swish_f(ds);
      }
    }
  }

  // ---- d_v per component, gated output
  for (int c = 0; c < 3; ++c) {
    const float* Av = agg_v + (size_t)mrowc * 384 + c;
    for (int nt = 0; nt < 4; ++nt) {
      v8f acc = zero8();
      for (int ks = 0; ks < 32; ++ks) {
        v2f a = afrag_row(Av, ks * 4, 3, lane);
        v2f b = bfrag(W_down_v, 64, ks * 4, nt * 16, lane);
        acc = wmma4(a, b, acc);
      }
      if (full) {
        for (int v = 0; v < 8; ++v) {
          int m = m0 + mb8 + v;
          float dv =
              acc[v] * CS + sc_v[(size_t)m * 192 + (nt * 16 + nloc) * 3 + c];
          out[(size_t)m * 256 + 64 + (nt * 16 + nloc) * 3 + c] =
              dv * gate[nt][v];
        }
      } else {
        for (int v = 0; v < 8; ++v) {
          int m = m0 + mb8 + v;
          if (m < N) {
            float dv =
                acc[v] * CS + sc_v[(size_t)m * 192 + (nt * 16 + nloc) * 3 + c];
            out[(size_t)m * 256 + 64 + (nt * 16 + nloc) * 3 + c] =
                dv * gate[nt][v];
          }
        }
      }
    }
  }
}

// ---------------------------------------------------------------------------
extern "C" void kernel_launch(void* const* d_in, const int* in_sizes, int n_in,
                              void* d_out, int out_size, void* d_ws,
                              size_t ws_size, hipStream_t stream) {
  const float* vectors = (const float*)d_in[0];
  const float* node_s = (const float*)d_in[1];
  const float* node_v = (const float*)d_in[2];
  const int* specie = (const int*)d_in[3];
  const int* senders = (const int*)d_in[4];
  const int* receivers = (const int*)d_in[5];
  const float* W_up_s = (const float*)d_in[6];
  const float* W_up_v = (const float*)d_in[7];
  const float* W_skip_s = (const float*)d_in[8];
  const float* W_skip_v = (const float*)d_in[9];
  const float* W1 = (const float*)d_in[10];
  const float* W2 = (const float*)d_in[11];
  const float* W3 = (const float*)d_in[12];
  const float* W_down_s = (const float*)d_in[13];
  const float* W_down_v = (const float*)d_in[14];
  float* out = (float*)d_out;

  const int N = in_sizes[1] / 64;  // node_scalars is [N,64]
  const int E = in_sizes[0] / 3;   // vectors is [E,3]

  // Workspace layout (floats): h_s, h_v, sc_s, sc_v, agg_s, agg_v
  float* ws = (float*)d_ws;
  size_t off = 0;
  float* h_s = ws + off;   off += (size_t)N * 64;
  float* h_v = ws + off;   off += (size_t)N * 192;
  float* sc_s = ws + off;  off += (size_t)N * 128;
  float* sc_v = ws + off;  off += (size_t)N * 192;
  float* agg_s = ws + off; off += (size_t)N * 128;
  float* agg_v = ws + off; off += (size_t)N * 384;

  // 1) zero aggregation buffers (agg_s and agg_v are contiguous)
  zero_kernel<<<2048, 256, 0, stream>>>(agg_s, (size_t)N * 512);

  // 2) node up + skip
  int nb_nodes = (N + 63) / 64;
  node_up_kernel<<<nb_nodes, 128, 0, stream>>>(
      node_s, node_v, specie, W_up_s, W_up_v, W_skip_s, W_skip_v, h_s, h_v,
      sc_s, sc_v, N);

  // 3) fused edge pipeline
  int nb_edges = (E + 31) / 32;
  edge_kernel<<<nb_edges, 64, 0, stream>>>(vectors, senders, receivers, W1, W2,
                                           W3, h_s, h_v, agg_s, agg_v, E);

  // 4) node down + gate
  node_down_kernel<<<nb_nodes, 128, 0, stream>>>(
      agg_s, agg_v, sc_s, sc_v, W_down_s, W_down_v, out, N);
}